// MeanAggregator_58514634441194
// MI455X (gfx1250) — compile-verified
//
#include <hip/hip_runtime.h>
#include <stdint.h>

// Problem constants (from the reference): features [N=200000, D=256] f32,
// nbrs [B=32768, S=10] int64, out [B, 256] f32.
#define NFEAT 256
#define NSAMP 10
#define WAVES_PER_BLOCK 4
#define THREADS_PER_BLOCK (WAVES_PER_BLOCK * 32)

// The gfx1250 async builtin takes v4i pointers in AS(1)/AS(3).
typedef int v4i_t __attribute__((__vector_size__(16)));
typedef __attribute__((address_space(1))) v4i_t g_v4i;
typedef __attribute__((address_space(3))) v4i_t l_v4i;

// Async global -> LDS copy of 16 bytes per lane (gfx1250 async-tensor path,
// tracked by ASYNCcnt).
__device__ __forceinline__ void async_gather_b128(const float* gptr, float* lptr) {
#if defined(__HIP_DEVICE_COMPILE__)
#if __has_builtin(__builtin_amdgcn_global_load_async_to_lds_b128)
  __builtin_amdgcn_global_load_async_to_lds_b128(
      (g_v4i*)(uintptr_t)(const void*)gptr,
      (l_v4i*)(uint32_t)(uintptr_t)(void*)lptr,
      /*offset=*/0, /*cpol=*/0);
#else
  uint32_t loff  = (uint32_t)(uintptr_t)(void*)lptr;        // LDS byte offset
  uint64_t gaddr = (uint64_t)(uintptr_t)(const void*)gptr;  // global address
  asm volatile("global_load_async_to_lds_b128 %0, %1, off"
               :: "v"(loff), "v"(gaddr) : "memory");
#endif
#endif
}

// Wait until at most N async ops remain outstanding (ASYNCcnt <= N).
// Async loads complete in order, so this gates exactly the oldest transfers.
template <int N>
__device__ __forceinline__ void wait_async_le() {
#if defined(__HIP_DEVICE_COMPILE__)
#if __has_builtin(__builtin_amdgcn_s_wait_asynccnt)
  __builtin_amdgcn_s_wait_asynccnt(N);
  asm volatile("" ::: "memory");  // keep LDS reads below the wait
#else
  asm volatile("s_wait_asynccnt %0" :: "i"(N) : "memory");
#endif
#endif
}

__global__ __launch_bounds__(THREADS_PER_BLOCK)
void mean_agg_gather_kernel(const float* __restrict__ features,
                            const long long* __restrict__ nbrs,
                            float* __restrict__ out,
                            int num_rows)
{
  // One wave (32 lanes) per output row; each wave owns a 10 KB LDS slab.
  __shared__ __align__(16) float smem[WAVES_PER_BLOCK][NSAMP][NFEAT];

  const int lane = (int)(threadIdx.x & 31u);
  // threadIdx.x>>5 is wave-invariant; readfirstlane makes that provable so the
  // nbrs row loads become scalar (s_load) and the de-dup runs on the SALU.
  const int wave = __builtin_amdgcn_readfirstlane((int)(threadIdx.x >> 5u));
  const int b    = (int)blockIdx.x * WAVES_PER_BLOCK + wave;
  if (b >= num_rows) return;  // no barriers in this kernel -> safe

  // ---- neighbor indices: wave-uniform -> scalar loads ----
  const long long* __restrict__ row = nbrs + (size_t)b * NSAMP;
  long long idx[NSAMP];
#pragma unroll
  for (int s = 0; s < NSAMP; ++s) idx[s] = row[s];

  // ---- kick off the async gather ASAP (20 coalesced 512B transfers) ----
  float* lslab = &smem[wave][0][0];
#pragma unroll
  for (int s = 0; s < NSAMP; ++s) {
    const float* g = features + (size_t)idx[s] * NFEAT + lane * 4;
    float*       l = lslab + s * NFEAT + lane * 4;
    async_gather_b128(g,       l);        // columns [0,128)
    async_gather_b128(g + 128, l + 128);  // columns [128,256)
  }

  // ---- de-dup weights while the gather is in flight (45 scalar compares) ----
  float w[NSAMP];
  int den = 0;
#pragma unroll
  for (int s = 0; s < NSAMP; ++s) {
    bool dup = false;
#pragma unroll
    for (int j = 0; j < NSAMP; ++j)
      if (j < s) dup = dup || (idx[j] == idx[s]);
    w[s] = dup ? 0.0f : 1.0f;
    den += dup ? 0 : 1;
  }

  // ---- pipelined accumulate: consume sample s once its two transfers landed
  // (in-order completion => ASYNCcnt <= 18-2s gates exactly sample s).
  float acc[8];
#pragma unroll
  for (int k = 0; k < 8; ++k) acc[k] = 0.0f;

#define CONSUME_SAMPLE(s_)                                                  \
  do {                                                                      \
    wait_async_le<2 * (NSAMP - 1 - (s_))>();                                \
    const float4* p = (const float4*)(lslab + (s_) * NFEAT + lane * 8);     \
    float4 a = p[0];                                                        \
    float4 c = p[1];                                                        \
    float ws = w[(s_)];                                                     \
    acc[0] = fmaf(ws, a.x, acc[0]);                                         \
    acc[1] = fmaf(ws, a.y, acc[1]);                                         \
    acc[2] = fmaf(ws, a.z, acc[2]);                                         \
    acc[3] = fmaf(ws, a.w, acc[3]);                                         \
    acc[4] = fmaf(ws, c.x, acc[4]);                                         \
    acc[5] = fmaf(ws, c.y, acc[5]);                                         \
    acc[6] = fmaf(ws, c.z, acc[6]);                                         \
    acc[7] = fmaf(ws, c.w, acc[7]);                                         \
  } while (0)

  CONSUME_SAMPLE(0);
  CONSUME_SAMPLE(1);
  CONSUME_SAMPLE(2);
  CONSUME_SAMPLE(3);
  CONSUME_SAMPLE(4);
  CONSUME_SAMPLE(5);
  CONSUME_SAMPLE(6);
  CONSUME_SAMPLE(7);
  CONSUME_SAMPLE(8);
  CONSUME_SAMPLE(9);
#undef CONSUME_SAMPLE

  const float inv = 1.0f / (float)den;  // den >= 1 always
  float4 o0 = make_float4(acc[0] * inv, acc[1] * inv, acc[2] * inv, acc[3] * inv);
  float4 o1 = make_float4(acc[4] * inv, acc[5] * inv, acc[6] * inv, acc[7] * inv);

  float4* op = (float4*)(out + (size_t)b * NFEAT + lane * 8);
  op[0] = o0;  // coalesced 1KB store per wave
  op[1] = o1;
}

extern "C" void kernel_launch(void* const* d_in, const int* in_sizes, int n_in,
                              void* d_out, int out_size, void* d_ws, size_t ws_size,
                              hipStream_t stream) {
  const float*     features = (const float*)d_in[0];      // f32 [N, 256]
  const long long* nbrs     = (const long long*)d_in[1];  // int64 [B, 10]
  float*           out      = (float*)d_out;              // f32 [B, 256]

  const int num_rows = in_sizes[1] / NSAMP;  // B = 32768
  const int blocks   = (num_rows + WAVES_PER_BLOCK - 1) / WAVES_PER_BLOCK;

  hipLaunchKernelGGL(mean_agg_gather_kernel,
                     dim3(blocks), dim3(THREADS_PER_BLOCK), 0, stream,
                     features, nbrs, out, num_rows);
}